// MinGRU_91070486544681
// MI455X (gfx1250) — compile-verified
//
#include <hip/hip_runtime.h>
#include <math.h>

// MinGRU: k = x@Wz^T+bz ; p = x@Wh^T+bh ; h_t = (1-sig(k))*h_{t-1} + sig(k)*g(p)
// Linear-space scan is exactly equivalent to the reference's log-space scan
// (all quantities positive, z in (0,1) -> convex combination, no overflow).

typedef __attribute__((ext_vector_type(2))) float v2f;
typedef __attribute__((ext_vector_type(8))) float v8f;

#define B_      8
#define T_      4096
#define D_      512
#define H_      512
#define M_      (B_ * T_)      // 32768 rows
#define CHUNKS  64
#define CS      64             // T_ / CHUNKS
#define MT      4              // 16-row M tiles per wave

// Branchless pair: returns sig_pos = sigmoid(|x|), sig_neg = sigmoid(-|x|)
// using a single v_exp_f32 + v_rcp_f32 (e = exp(-|x|) <= 1, never overflows).
__device__ __forceinline__ void sigmoid_pair(float x, float& sig_pos, float& sig_neg) {
    const float e = __expf(-fabsf(x));
    const float r = __builtin_amdgcn_rcpf(1.0f + e);
    sig_pos = r;          // sigmoid(+|x|)
    sig_neg = e * r;      // sigmoid(-|x|)
}

// g(x) = x + 0.5 if x >= 0 else sigmoid(x). Branchless, overflow-safe.
__device__ __forceinline__ float g_fn(float x) {
    const float e = __expf(fminf(x, 0.0f));                 // exp(x) for x<0, 1 otherwise
    const float s = e * __builtin_amdgcn_rcpf(1.0f + e);    // sigmoid(x) for x<0
    return (x >= 0.0f) ? (x + 0.5f) : s;
}

// ---------------------------------------------------------------------------
// Phase 1: dual GEMM via V_WMMA_F32_16X16X4_F32 + fused gate epilogue.
// One wave per workgroup; wave computes a 64(M) x 16(N) tile of BOTH GEMMs,
// sharing the A (activation) loads across the Wz and Wh accumulator chains.
// Writes a = 1 - sigmoid(k)  and  v = sigmoid(k) * g(p)  so the scan phases
// are pure FMA streams.
// ---------------------------------------------------------------------------
__global__ __launch_bounds__(32) void mingru_gemm_gates(
    const float* __restrict__ X,
    const float* __restrict__ Wz, const float* __restrict__ bz,
    const float* __restrict__ Wh, const float* __restrict__ bh,
    float* __restrict__ outA, float* __restrict__ outV)
{
    const int lane = threadIdx.x;        // wave32
    const int half = lane >> 4;          // 0: K=0,1   1: K=2,3
    const int l16  = lane & 15;
    const int m0   = blockIdx.x * (16 * MT);
    const int n0   = blockIdx.y * 16;

    // A-matrix (16x4 f32): lane = M row, VGPR0/1 = K pair (contiguous float2)
    const float* aptr[MT];
#pragma unroll
    for (int mt = 0; mt < MT; ++mt)
        aptr[mt] = X + (size_t)(m0 + mt * 16 + l16) * D_ + half * 2;
    // B-matrix (4x16 f32): lane = N column (W row, row-major (H,D)), same K pair
    const float* bzptr = Wz + (size_t)(n0 + l16) * D_ + half * 2;
    const float* bhptr = Wh + (size_t)(n0 + l16) * D_ + half * 2;

    v8f accZ[MT], accH[MT];
#pragma unroll
    for (int mt = 0; mt < MT; ++mt) {
        accZ[mt] = (v8f){0.f,0.f,0.f,0.f,0.f,0.f,0.f,0.f};
        accH[mt] = (v8f){0.f,0.f,0.f,0.f,0.f,0.f,0.f,0.f};
    }

#pragma unroll 2
    for (int k = 0; k < D_; k += 4) {
        v2f bzv = *(const v2f*)(bzptr + k);
        v2f bhv = *(const v2f*)(bhptr + k);
#pragma unroll
        for (int mt = 0; mt < MT; ++mt) {
            v2f av = *(const v2f*)(aptr[mt] + k);
            accZ[mt] = __builtin_amdgcn_wmma_f32_16x16x4_f32(
                false, av, false, bzv, (short)0, accZ[mt], false, false);
            accH[mt] = __builtin_amdgcn_wmma_f32_16x16x4_f32(
                false, av, false, bhv, (short)0, accH[mt], false, false);
        }
    }

    // Epilogue. C/D layout: VGPR r, lane<16 -> (M=r, N=lane); lane>=16 -> (M=8+r, N=lane-16)
    const int n = n0 + l16;
    const float bzn = bz[n];
    const float bhn = bh[n];
#pragma unroll
    for (int mt = 0; mt < MT; ++mt) {
#pragma unroll
        for (int r = 0; r < 8; ++r) {
            const int m   = m0 + mt * 16 + half * 8 + r;
            const float kv  = accZ[mt][r] + bzn;
            const float pre = accH[mt][r] + bhn;
            float sp, sn;
            sigmoid_pair(kv, sp, sn);
            const float z = (kv >= 0.0f) ? sp : sn;   // sigmoid(kv)
            const float a = (kv >= 0.0f) ? sn : sp;   // sigmoid(-kv) = 1 - z
            const float v = z * g_fn(pre);
            const size_t idx = (size_t)m * H_ + n;
            outA[idx] = a;
            outV[idx] = v;
        }
    }
}

// ---------------------------------------------------------------------------
// Phase 2a: per-chunk affine reduction. h_out = A*h_in + V over 64 steps.
// 262144 threads, h-contiguous (coalesced) streaming of a/v.
// ---------------------------------------------------------------------------
__global__ __launch_bounds__(256) void mingru_chunk_reduce(
    const float* __restrict__ a, const float* __restrict__ v,
    float* __restrict__ Ac, float* __restrict__ Vc)
{
    const int tid = blockIdx.x * 256 + threadIdx.x;   // (b*CHUNKS + c)*H + h
    const int h  = tid % H_;
    const int bc = tid / H_;
    const int c  = bc % CHUNKS;
    const int b  = bc / CHUNKS;
    size_t base = ((size_t)(b * T_ + c * CS)) * H_ + h;

    float A = 1.0f, V = 0.0f;
#pragma unroll 4
    for (int t = 0; t < CS; ++t) {
        const float at = a[base];
        const float vt = v[base];
        A = A * at;
        V = fmaf(at, V, vt);
        base += H_;
    }
    Ac[tid] = A;
    Vc[tid] = V;
}

// ---------------------------------------------------------------------------
// Phase 2b: tiny sequential scan over the 64 chunk summaries per channel.
// ---------------------------------------------------------------------------
__global__ __launch_bounds__(256) void mingru_chunk_scan(
    const float* __restrict__ h0,
    const float* __restrict__ Ac, const float* __restrict__ Vc,
    float* __restrict__ hin)
{
    const int tid = blockIdx.x * 256 + threadIdx.x;   // b*H + h
    const int h = tid % H_;
    const int b = tid / H_;
    float hs = g_fn(h0[(size_t)b * H_ + h]);          // exp(log_g(h0))
#pragma unroll 8
    for (int c = 0; c < CHUNKS; ++c) {
        const size_t i = ((size_t)(b * CHUNKS + c)) * H_ + h;
        hin[i] = hs;
        hs = fmaf(Ac[i], hs, Vc[i]);
    }
}

// ---------------------------------------------------------------------------
// Phase 2c: replay each chunk with its correct incoming state; write output.
// ---------------------------------------------------------------------------
__global__ __launch_bounds__(256) void mingru_chunk_apply(
    const float* __restrict__ a, const float* __restrict__ v,
    const float* __restrict__ hin, float* __restrict__ out)
{
    const int tid = blockIdx.x * 256 + threadIdx.x;
    const int h  = tid % H_;
    const int bc = tid / H_;
    const int c  = bc % CHUNKS;
    const int b  = bc / CHUNKS;
    size_t base = ((size_t)(b * T_ + c * CS)) * H_ + h;

    float hs = hin[tid];
#pragma unroll 4
    for (int t = 0; t < CS; ++t) {
        hs = fmaf(a[base], hs, v[base]);
        out[base] = hs;
        base += H_;
    }
}

// ---------------------------------------------------------------------------
// Workspace: a (M*H) | v (M*H) | Ac (B*CHUNKS*H) | Vc | hin  ~= 131 MB
// ---------------------------------------------------------------------------
extern "C" void kernel_launch(void* const* d_in, const int* in_sizes, int n_in,
                              void* d_out, int out_size, void* d_ws, size_t ws_size,
                              hipStream_t stream) {
    const float* x  = (const float*)d_in[0];
    const float* h0 = (const float*)d_in[1];
    const float* Wz = (const float*)d_in[2];
    const float* bz = (const float*)d_in[3];
    const float* Wh = (const float*)d_in[4];
    const float* bh = (const float*)d_in[5];
    float* out = (float*)d_out;

    float* wa  = (float*)d_ws;
    float* wv  = wa + (size_t)M_ * H_;
    float* Ac  = wv + (size_t)M_ * H_;
    float* Vc  = Ac + (size_t)B_ * CHUNKS * H_;
    float* hin = Vc + (size_t)B_ * CHUNKS * H_;

    dim3 g1(M_ / (16 * MT), H_ / 16);
    mingru_gemm_gates<<<g1, 32, 0, stream>>>(x, Wz, bz, Wh, bh, wa, wv);

    const int nchunk = B_ * H_ * CHUNKS;             // 262144
    mingru_chunk_reduce<<<nchunk / 256, 256, 0, stream>>>(wa, wv, Ac, Vc);
    mingru_chunk_scan<<<(B_ * H_) / 256, 256, 0, stream>>>(h0, Ac, Vc, hin);
    mingru_chunk_apply<<<nchunk / 256, 256, 0, stream>>>(wa, wv, hin, out);
}